// YesTF_GradKP_DetectionConfidenceMap2keypoint_43602507989182
// MI455X (gfx1250) — compile-verified
//
#include <hip/hip_runtime.h>
#include <cstdint>

// One workgroup (256 threads = 8 wave32) per channel. 1024 channels per
// tensor, 2 tensors -> 2048 blocks. Channel = 16384 floats = 64KB, staged
// once into LDS via gfx1250 async global->LDS loads (ASYNCcnt path), then
// both the reduction pass and the normalize pass read LDS only.
__global__ __launch_bounds__(256) void hm2kp_kernel(const float* __restrict__ in0,
                                                    const float* __restrict__ in1,
                                                    float* __restrict__ out)
{
    constexpr int    HW     = 128 * 128;          // 16384 elements/channel
    constexpr int    VPT    = HW / 4 / 256;       // 16 float4 per thread
    constexpr size_t O_KP0  = 16777216;           // after map_val_all [B,C,H,W]
    constexpr size_t O_ZETA = O_KP0 + 2048;       // after keypoint   [B,C,2]
    constexpr size_t O_MAP1 = O_ZETA + 1024;      // after zeta       [B,C]
    constexpr size_t O_KP1  = O_MAP1 + 16777216;  // after tf map

    __shared__ float s_data[HW];        // 64 KB channel tile
    __shared__ float s_part[5][8];      // per-wave partials: mn,mx,s,sx,sy
    __shared__ float s_fin[2];          // broadcast: mn, 1/(mx-mn)

    const uint32_t tid   = threadIdx.x;
    const uint32_t blk   = blockIdx.x;
    const uint32_t which = blk >> 10;   // 0 = combined_hm_preds, 1 = tf_*
    const uint32_t ch    = blk & 1023;  // flat b*C + c

    const float* src = (which ? in1 : in0) + (size_t)ch * HW;
    float* out_map   = out + (which ? O_MAP1 : (size_t)0) + (size_t)ch * HW;
    float* out_kp    = out + (which ? O_KP1 : O_KP0) + (size_t)ch * 2u;

    // addrspace(3) offset = low 32 bits of the generic shared pointer
    const uint32_t lds_base = (uint32_t)(uintptr_t)(&s_data[0]);
    const unsigned long long gbase = (unsigned long long)(uintptr_t)src;

    // ---- Phase 0: async copy global -> LDS (CDNA5 TDM-family async path) ----
    #pragma unroll
    for (int k = 0; k < VPT; ++k) {
        const uint32_t goff  = (tid + (uint32_t)(k << 8)) << 4; // byte offset of float4
        const uint32_t laddr = lds_base + goff;
        asm volatile("global_load_async_to_lds_b128 %0, %1, %2"
                     :: "v"(laddr), "v"(goff), "s"(gbase) : "memory");
    }
    // Each lane reads only the LDS bytes it loaded itself -> per-wave wait suffices.
    asm volatile("s_wait_asynccnt 0" ::: "memory");

    // ---- Phase 1: per-thread reductions on raw data ----
    float mn = __builtin_inff();
    float mx = -__builtin_inff();
    float s = 0.f, sx = 0.f, sy = 0.f;   // S = sum R, Sx = sum w*R, Sy = sum h*R
    #pragma unroll
    for (int k = 0; k < VPT; ++k) {
        const int e4 = (int)tid + (k << 8);
        const float4 v = *reinterpret_cast<const float4*>(&s_data[e4 << 2]);
        const int e = e4 << 2;                 // element index within channel
        const float h  = (float)(e >> 7);      // row (W = 128)
        const float w0 = (float)(e & 127);     // first column of this float4
        const float rs = (v.x + v.y) + (v.z + v.w);
        s  += rs;
        sy += h * rs;
        sx += w0 * rs + (v.y + 2.0f * v.z + 3.0f * v.w);
        mn = fminf(mn, fminf(fminf(v.x, v.y), fminf(v.z, v.w)));
        mx = fmaxf(mx, fmaxf(fmaxf(v.x, v.y), fmaxf(v.z, v.w)));
    }

    // wave32 butterfly reduction
    #pragma unroll
    for (int o = 16; o > 0; o >>= 1) {
        mn = fminf(mn, __shfl_xor(mn, o, 32));
        mx = fmaxf(mx, __shfl_xor(mx, o, 32));
        s  += __shfl_xor(s,  o, 32);
        sx += __shfl_xor(sx, o, 32);
        sy += __shfl_xor(sy, o, 32);
    }
    const uint32_t wave = tid >> 5;
    if ((tid & 31u) == 0u) {
        s_part[0][wave] = mn; s_part[1][wave] = mx;
        s_part[2][wave] = s;  s_part[3][wave] = sx; s_part[4][wave] = sy;
    }
    __syncthreads();

    if (tid == 0) {
        mn = s_part[0][0]; mx = s_part[1][0];
        s  = s_part[2][0]; sx = s_part[3][0]; sy = s_part[4][0];
        #pragma unroll
        for (int wv = 1; wv < 8; ++wv) {
            mn = fminf(mn, s_part[0][wv]);
            mx = fmaxf(mx, s_part[1][wv]);
            s  += s_part[2][wv]; sx += s_part[3][wv]; sy += s_part[4][wv];
        }
        const float inv  = 1.0f / (mx - mn);
        // map = (R - mn)*inv  =>  sums over map derived from raw sums:
        const float zeta = (s - 16384.0f * mn) * inv;          // N = H*W
        const float csum = 1040384.0f;                         // H * (W-1)*W/2 (== W*(H-1)*H/2)
        const float kpx  = (sx - csum * mn) * inv;
        const float kpy  = (sy - csum * mn) * inv;
        out_kp[0] = rintf(kpx / zeta);                         // jnp.round = half-even
        out_kp[1] = rintf(kpy / zeta);
        if (which == 0u) out[O_ZETA + ch] = zeta;
        s_fin[0] = mn; s_fin[1] = inv;
    }
    __syncthreads();

    // ---- Phase 2: normalize from LDS, stream out coalesced float4 ----
    mn = s_fin[0];
    const float inv = s_fin[1];
    #pragma unroll
    for (int k = 0; k < VPT; ++k) {
        const int e4 = (int)tid + (k << 8);
        const float4 v = *reinterpret_cast<const float4*>(&s_data[e4 << 2]);
        float4 r;
        r.x = (v.x - mn) * inv; r.y = (v.y - mn) * inv;
        r.z = (v.z - mn) * inv; r.w = (v.w - mn) * inv;
        *reinterpret_cast<float4*>(&out_map[e4 << 2]) = r;
    }
}

extern "C" void kernel_launch(void* const* d_in, const int* in_sizes, int n_in,
                              void* d_out, int out_size, void* d_ws, size_t ws_size,
                              hipStream_t stream) {
    const float* in0 = (const float*)d_in[0];  // combined_hm_preds   [16,64,128,128] f32
    const float* in1 = (const float*)d_in[1];  // tf_combined_hm_preds
    float* outp = (float*)d_out;
    (void)in_sizes; (void)n_in; (void)out_size; (void)d_ws; (void)ws_size;
    hm2kp_kernel<<<dim3(2048), dim3(256), 0, stream>>>(in0, in1, outp);
}